// GuidedFilter_35931696398902
// MI455X (gfx1250) — compile-verified
//
#include <hip/hip_runtime.h>
#include <hip/hip_bf16.h>
#include <stdint.h>

typedef float v2f __attribute__((ext_vector_type(2)));
typedef float v8f __attribute__((ext_vector_type(8)));

#define HH 1024
#define WW 1024
#define NPLANE 24          // 8 batch * 3 channels
#define TILE 32
#define GF_EPS 1e-6f

__device__ __forceinline__ int iclamp(int v, int lo, int hi) {
    return v < lo ? lo : (v > hi ? hi : v);
}

// CDNA5 ISA 10.2: for generic pointers in the LDS aperture, the LDS address is the low 32 bits of the address.
__device__ __forceinline__ unsigned lds_addr(const void* p) {
    return (unsigned)(unsigned long long)p;
}

// Async DMA: global -> LDS, tracked by ASYNCcnt (no VGPR staging).
__device__ __forceinline__ void async_load_f32(void* lds_dst, const float* gsrc) {
    asm volatile("global_load_async_to_lds_b32 %0, %1, off"
                 :: "v"(lds_addr(lds_dst)), "v"(gsrc)
                 : "memory");
}

__global__ __launch_bounds__(128) void guided_filter_fused(
    const float* __restrict__ x, const float* __restrict__ y,
    float* __restrict__ out)
{
    // x/y tiles with halo 2 (rows/cols -2..33), padded stride vs bank conflicts
    __shared__ float xs[36][40];
    __shared__ float ys[36][40];
    // A, b on halo-1 region (tile rows/cols -1..32): 34x34, stride 36
    __shared__ float Aa[34][36];
    __shared__ float Ab[34][36];
    // Horizontal 3-sums of A, b: rows -1..32 (34), cols 0..31
    __shared__ float Qa[34][32];
    __shared__ float Qb[34][32];

    const int  tid = threadIdx.x;
    const int  gx0 = blockIdx.x * TILE;
    const int  gy0 = blockIdx.y * TILE;
    const long plane = (long)blockIdx.z * (long)(HH * WW);
    const float* xp = x + plane;
    const float* yp = y + plane;

    // ---- Stage 0: async tile load, replication pad via coordinate clamp ----
    for (int i = tid; i < 36 * 36; i += 128) {
        int r = i / 36, c = i - r * 36;
        int gr = iclamp(gy0 + r - 2, 0, HH - 1);
        int gc = iclamp(gx0 + c - 2, 0, WW - 1);
        const float* sx = xp + (long)gr * WW + gc;
        const float* sy = yp + (long)gr * WW + gc;
        async_load_f32(&xs[r][c], sx);
        async_load_f32(&ys[r][c], sy);
    }
    asm volatile("s_wait_asynccnt 0" ::: "memory");
    __syncthreads();

    const float inv9 = 1.0f / 9.0f;

    // ---- Stage 1: A, b on the 34x34 halo-1 region ----
    // Clamp the CENTER first so halo A/b equal the replication-padded values.
    for (int i = tid; i < 34 * 34; i += 128) {
        int r = i / 34, c = i - r * 34;           // array idx; tile coords r-1, c-1
        int gyc = iclamp(gy0 + r - 1, 0, HH - 1);
        int gxc = iclamp(gx0 + c - 1, 0, WW - 1);
        float sx = 0.f, sy = 0.f, sxx = 0.f, sxy = 0.f;
        #pragma unroll
        for (int dy = -1; dy <= 1; ++dy) {
            #pragma unroll
            for (int dx = -1; dx <= 1; ++dx) {
                int rr = iclamp(gyc + dy, 0, HH - 1) - gy0 + 2;
                int cc = iclamp(gxc + dx, 0, WW - 1) - gx0 + 2;
                float xv = xs[rr][cc];
                float yv = ys[rr][cc];
                sx += xv; sy += yv; sxx += xv * xv; sxy += xv * yv;
            }
        }
        float mx  = sx * inv9, my = sy * inv9;
        float cov = sxy * inv9 - mx * my;
        float var = sxx * inv9 - mx * mx;
        float Av  = cov / (var + GF_EPS);
        float Bv  = my - Av * mx;
        Aa[r][c] = Av;
        Ab[r][c] = Bv;
    }
    __syncthreads();

    // ---- Stage 2a: horizontal 3-sums (cheap on VALU) ----
    for (int i = tid; i < 34 * 32; i += 128) {
        int r = i >> 5, c = i & 31;
        Qa[r][c] = Aa[r][c] + Aa[r][c + 1] + Aa[r][c + 2];
        Qb[r][c] = Ab[r][c] + Ab[r][c + 1] + Ab[r][c + 2];
    }
    __syncthreads();

    // ---- Stage 2b: vertical box as banded matmul on the matrix pipe ----
    // Out(16x16) = V(16x20) * Q(20x16), V[m][k] = 1/9 * (m <= k <= m+2, k<18).
    const int wave = tid >> 5;
    const int lane = tid & 31;
    const int m    = lane & 15;
    const int hi   = lane >> 4;
    const int sr   = (wave >> 1) * 16;   // sub-tile row origin (0 or 16)
    const int sc   = (wave & 1) * 16;    // sub-tile col origin (0 or 16)

    v8f cA = {};
    v8f cB = {};

#if defined(__gfx1250__) && __has_builtin(__builtin_amdgcn_wmma_f32_16x16x4_f32)
    #pragma unroll
    for (int k = 0; k < 5; ++k) {
        // ISA layout for 32-bit A(16x4) / B(4x16): lanes 0-15 hold K pair {4k,4k+1},
        // lanes 16-31 hold {4k+2,4k+3}; A indexed by M=m, B by N=m.
        int kk0 = 4 * k + 2 * hi;
        int kk1 = kk0 + 1;
        v2f a;
        a.x = (kk0 >= m && kk0 <= m + 2 && kk0 < 18) ? inv9 : 0.0f;
        a.y = (kk1 >= m && kk1 <= m + 2 && kk1 < 18) ? inv9 : 0.0f;
        int r0 = sr + (kk0 < 18 ? kk0 : 17);  // pad rows carry weight 0
        int r1 = sr + (kk1 < 18 ? kk1 : 17);
        v2f ba; ba.x = Qa[r0][sc + m]; ba.y = Qa[r1][sc + m];
        v2f bb; bb.x = Qb[r0][sc + m]; bb.y = Qb[r1][sc + m];
        cA = __builtin_amdgcn_wmma_f32_16x16x4_f32(false, a, false, ba,
                                                   (short)0, cA, false, false);
        cB = __builtin_amdgcn_wmma_f32_16x16x4_f32(false, a, false, bb,
                                                   (short)0, cB, false, false);
    }
#else
    // VALU fallback, identical math, same C layout (VGPR v <-> M = v + 8*hi).
    #pragma unroll
    for (int v = 0; v < 8; ++v) {
        int R  = sr + v + hi * 8;
        int Cc = sc + m;
        cA[v] = (Qa[R][Cc] + Qa[R + 1][Cc] + Qa[R + 2][Cc]) * inv9;
        cB[v] = (Qb[R][Cc] + Qb[R + 1][Cc] + Qb[R + 2][Cc]) * inv9;
    }
#endif

    // ---- Epilogue: res = box(A)*x + box(b), trunc, clamp, NT store ----
    float* op = out + plane;
    #pragma unroll
    for (int v = 0; v < 8; ++v) {
        int R  = sr + v + hi * 8;      // C layout: VGPR v, lanes 16-31 -> M = v+8
        int Cc = sc + m;
        float xv  = xs[R + 2][Cc + 2];
        float res = cA[v] * xv + cB[v];
        float t   = truncf(res);
        t = t < 0.0f ? 0.0f : (t > 255.0f ? 255.0f : t);
        __builtin_nontemporal_store(t, op + (long)(gy0 + R) * WW + (gx0 + Cc));
    }
}

extern "C" void kernel_launch(void* const* d_in, const int* in_sizes, int n_in,
                              void* d_out, int out_size, void* d_ws, size_t ws_size,
                              hipStream_t stream) {
    (void)in_sizes; (void)n_in; (void)out_size; (void)d_ws; (void)ws_size;
    const float* x = (const float*)d_in[0];
    const float* y = (const float*)d_in[1];
    float* out = (float*)d_out;
    dim3 grid(WW / TILE, HH / TILE, NPLANE);
    guided_filter_fused<<<grid, 128, 0, stream>>>(x, y, out);
}